// SSIMLoss2D_MC_16655883174397
// MI455X (gfx1250) — compile-verified
//
#include <hip/hip_runtime.h>
#include <stdint.h>

// ---------------------------------------------------------------------------
// SSIM loss, (32,8,320,320) f32, 7x7 uniform box window.
// BW-bound: ~210 MB -> ~9us floor at 23.3 TB/s. TDM (tensor_load_to_lds)
// stages strips into LDS; float4 (b128) sliding-window separable box sums;
// quad-output horizontal pass; deterministic two-stage reduction.
// ---------------------------------------------------------------------------

#define IMG      320
#define WIN      7
#define OUT_W    314                        // IMG - WIN + 1
#define OUT_H    12                         // output rows per strip
#define IN_H     (OUT_H + WIN - 1)          // 18 input rows per strip
#define STRIPS   ((OUT_W + OUT_H - 1) / OUT_H)  // 27
#define NTHREADS 160                        // 5 waves (wave32)
#define NQUAD    (IMG / 4)                  // 80 quad-columns
#define PADQ     82                         // padded V-plane width in quads
#define COVN     (49.0f / 48.0f)            // NP_WIN / (NP_WIN - 1)

typedef unsigned int       u32;
typedef unsigned long long u64;
typedef u32 v4u __attribute__((ext_vector_type(4)));
typedef int v8i __attribute__((ext_vector_type(8)));
typedef int v4i __attribute__((ext_vector_type(4)));

#if defined(__gfx1250__)
// Minimal 2D TDM descriptor (D#): group0 = count/lds_addr/global_addr/type,
// group1 = data_size, tensor dims (OOB zero-fill), tile dims, dim0 stride.
__device__ __forceinline__ void tdm_load_tile_2d(u32 lds_byte_addr, u64 gaddr,
                                                 u32 tile_w, u32 tile_h,
                                                 u32 tensor_w, u32 tensor_h,
                                                 u32 row_stride_elems) {
  v4u g0;
  g0.x = 1u;                                            // count=1 (valid D#)
  g0.y = lds_byte_addr;                                 // LDS byte address
  g0.z = (u32)(gaddr & 0xFFFFFFFFull);                  // global_addr[31:0]
  g0.w = (u32)((gaddr >> 32) & 0x01FFFFFFull)           // global_addr[56:32]
       | (2u << 30);                                    // type=2 ("image")
  v8i g1;
  g1[0] = (int)(2u << 16);                              // data_size=2 -> 4 bytes
  g1[1] = (int)((tensor_w & 0xFFFFu) << 16);            // tensor_dim0[15:0]
  g1[2] = (int)(((tensor_w >> 16) & 0xFFFFu)
              | ((tensor_h & 0xFFFFu) << 16));          // dim0[31:16] | dim1[15:0]
  g1[3] = (int)(((tensor_h >> 16) & 0xFFFFu)
              | ((tile_w & 0xFFFFu) << 16));            // dim1[31:16] | tile_dim0
  g1[4] = (int)(tile_h & 0xFFFFu);                      // tile_dim1 (tile_dim2=0)
  g1[5] = (int)row_stride_elems;                        // tensor_dim0_stride[31:0]
  g1[6] = 0;
  g1[7] = 0;
  v4i z4 = {0, 0, 0, 0};
#if __clang_major__ >= 23
  v8i z8 = {0, 0, 0, 0, 0, 0, 0, 0};
  __builtin_amdgcn_tensor_load_to_lds(g0, g1, z4, z4, z8, 0);
#else
  __builtin_amdgcn_tensor_load_to_lds(g0, g1, z4, z4, 0);
#endif
}
#endif

__global__ __launch_bounds__(NTHREADS)
void ssim_strip_kernel(const float* __restrict__ X, const float* __restrict__ Y,
                       const float* __restrict__ drp, const float* __restrict__ wp,
                       float* __restrict__ partial) {
  __shared__ float4 sX4[IN_H * NQUAD];      // 23,040 B
  __shared__ float4 sY4[IN_H * NQUAD];      // 23,040 B
  __shared__ float4 sV4[2][5][PADQ];        // 13,120 B (2 rows x 5 qty)
  __shared__ float  sRed[NTHREADS];         //    640 B  -> 59,840 B total

  const int bx  = blockIdx.x;
  const int map = bx / STRIPS;
  const int st  = bx % STRIPS;
  const int r0  = st * OUT_H;
  const int orows  = min(OUT_H, OUT_W - r0);
  const int inrows = orows + WIN - 1;
  const u64 elem_off = (u64)map * (u64)(IMG * IMG) + (u64)r0 * (u64)IMG;

#if defined(__gfx1250__)
  if (threadIdx.x < 32u) {   // one wave issues both DMA descriptors
    tdm_load_tile_2d((u32)(uintptr_t)(&sX4[0]),
                     (u64)(uintptr_t)X + 4ull * elem_off,
                     (u32)IMG, (u32)inrows, (u32)IMG, (u32)(IMG - r0), (u32)IMG);
    tdm_load_tile_2d((u32)(uintptr_t)(&sY4[0]),
                     (u64)(uintptr_t)Y + 4ull * elem_off,
                     (u32)IMG, (u32)inrows, (u32)IMG, (u32)(IMG - r0), (u32)IMG);
    __builtin_amdgcn_s_wait_tensorcnt(0);
  }
#else
  {
    const float4* Xg = (const float4*)(X + elem_off);
    const float4* Yg = (const float4*)(Y + elem_off);
    for (int i = threadIdx.x; i < inrows * NQUAD; i += NTHREADS) {
      sX4[i] = Xg[i];
      sY4[i] = Yg[i];
    }
  }
#endif
  __syncthreads();

  const float w0 = wp[0];                // uniform box weight (1/49)
  const float dr = drp[0];
  const float C1 = (0.01f * dr) * (0.01f * dr);
  const float C2 = (0.03f * dr) * (0.03f * dr);
  const int tid = threadIdx.x;

  // ---- vertical running sums: thread q owns quad-column q (q < 80) ----
  float4 ax = {0,0,0,0}, ay = {0,0,0,0}, axx = {0,0,0,0},
         ayy = {0,0,0,0}, axy = {0,0,0,0};

#define VADD(r)  do { \
    const float4 x_ = sX4[(r) * NQUAD + tid]; \
    const float4 y_ = sY4[(r) * NQUAD + tid]; \
    ax.x += x_.x;        ax.y += x_.y;        ax.z += x_.z;        ax.w += x_.w; \
    ay.x += y_.x;        ay.y += y_.y;        ay.z += y_.z;        ay.w += y_.w; \
    axx.x += x_.x*x_.x;  axx.y += x_.y*x_.y;  axx.z += x_.z*x_.z;  axx.w += x_.w*x_.w; \
    ayy.x += y_.x*y_.x;  ayy.y += y_.y*y_.y;  ayy.z += y_.z*y_.z;  ayy.w += y_.w*y_.w; \
    axy.x += x_.x*y_.x;  axy.y += x_.y*y_.y;  axy.z += x_.z*y_.z;  axy.w += x_.w*y_.w; \
  } while (0)

#define VSUB(r)  do { \
    const float4 x_ = sX4[(r) * NQUAD + tid]; \
    const float4 y_ = sY4[(r) * NQUAD + tid]; \
    ax.x -= x_.x;        ax.y -= x_.y;        ax.z -= x_.z;        ax.w -= x_.w; \
    ay.x -= y_.x;        ay.y -= y_.y;        ay.z -= y_.z;        ay.w -= y_.w; \
    axx.x -= x_.x*x_.x;  axx.y -= x_.y*x_.y;  axx.z -= x_.z*x_.z;  axx.w -= x_.w*x_.w; \
    ayy.x -= y_.x*y_.x;  ayy.y -= y_.y*y_.y;  ayy.z -= y_.z*y_.z;  ayy.w -= y_.w*y_.w; \
    axy.x -= x_.x*y_.x;  axy.y -= x_.y*y_.y;  axy.z -= x_.z*y_.z;  axy.w -= x_.w*y_.w; \
  } while (0)

#define VPUB(s)  do { \
    sV4[s][0][tid] = ax;  sV4[s][1][tid] = ay;  sV4[s][2][tid] = axx; \
    sV4[s][3][tid] = ayy; sV4[s][4][tid] = axy; \
  } while (0)

  if (tid < NQUAD) {
    for (int r = 0; r < WIN - 1; ++r) VADD(r);   // prime rows 0..5
  }

  float acc = 0.0f;
  for (int rr = 0; rr < orows; rr += 2) {
    // Publish vertical sums for rows rr and rr+1.
    if (tid < NQUAD) {
      VADD(rr + WIN - 1);
      VPUB(0);
      if (rr + 1 < orows) { VADD(rr + WIN); VPUB(1); }
    }
    __syncthreads();

    // Horizontal 7-tap (quad sliding) + SSIM: 160 items = 2 rows x 80 quads.
    {
      const int q = tid % NQUAD;       // quad-column
      const int r = tid / NQUAD;       // row slot 0/1
      const int c0 = q * 4;
      if ((rr + r) < orows && c0 < OUT_W) {
        float4 h[5];
#pragma unroll
        for (int k = 0; k < 5; ++k) {
          const float4 v0 = sV4[r][k][q];
          const float4 v1 = sV4[r][k][q + 1];
          const float4 v2 = sV4[r][k][q + 2];
          float s = ((v0.x + v0.y) + (v0.z + v0.w)) + ((v1.x + v1.y) + v1.z);
          h[k].x = s;
          s += v1.w - v0.x;  h[k].y = s;
          s += v2.x - v0.y;  h[k].z = s;
          s += v2.y - v0.z;  h[k].w = s;
        }
        const float* h0 = (const float*)&h[0];
        const float* h1 = (const float*)&h[1];
        const float* h2 = (const float*)&h[2];
        const float* h3 = (const float*)&h[3];
        const float* h4 = (const float*)&h[4];
#pragma unroll
        for (int j = 0; j < 4; ++j) {
          const float ux = h0[j] * w0, uy = h1[j] * w0;
          const float uxx = h2[j] * w0, uyy = h3[j] * w0, uxy = h4[j] * w0;
          const float vx  = COVN * (uxx - ux * ux);
          const float vy  = COVN * (uyy - uy * uy);
          const float vxy = COVN * (uxy - ux * uy);
          const float A1 = 2.0f * ux * uy + C1;
          const float A2 = 2.0f * vxy + C2;
          const float B1 = ux * ux + uy * uy + C1;
          const float B2 = vx + vy + C2;
          const float S = (A1 * A2) / (B1 * B2 + 1e-8f);
          acc += (c0 + j < OUT_W) ? S : 0.0f;
        }
      }
    }
    __syncthreads();

    // Retire rows rr (and rr+1) from the vertical window.
    if (tid < NQUAD) {
      VSUB(rr);
      if (rr + 1 < orows) VSUB(rr + 1);
    }
  }

  // Deterministic block reduction -> per-block partial sum.
  sRed[tid] = acc;
  __syncthreads();
  for (int s = 128; s > 0; s >>= 1) {
    if (tid < s && tid + s < NTHREADS) sRed[tid] += sRed[tid + s];
    __syncthreads();
  }
  if (tid == 0) partial[bx] = sRed[0];
}

__global__ __launch_bounds__(256)
void ssim_finalize_kernel(const float* __restrict__ partial, int n,
                          double inv_total, float* __restrict__ out) {
  __shared__ double sred[256];
  double a = 0.0;
  for (int i = threadIdx.x; i < n; i += 256) a += (double)partial[i];
  sred[threadIdx.x] = a;
  __syncthreads();
  for (int s = 128; s > 0; s >>= 1) {
    if ((int)threadIdx.x < s) sred[threadIdx.x] += sred[threadIdx.x + s];
    __syncthreads();
  }
  if (threadIdx.x == 0) out[0] = 1.0f - (float)(sred[0] * inv_total);
}

extern "C" void kernel_launch(void* const* d_in, const int* in_sizes, int n_in,
                              void* d_out, int out_size, void* d_ws, size_t ws_size,
                              hipStream_t stream) {
  (void)n_in; (void)out_size; (void)ws_size;
  const float* X  = (const float*)d_in[0];
  const float* Y  = (const float*)d_in[1];
  const float* dr = (const float*)d_in[2];
  const float* w  = (const float*)d_in[3];
  float* out      = (float*)d_out;
  float* partial  = (float*)d_ws;   // nblocks floats of scratch

  const int maps    = in_sizes[0] / (IMG * IMG);   // 32*8 = 256
  const int nblocks = maps * STRIPS;               // 6912

  ssim_strip_kernel<<<nblocks, NTHREADS, 0, stream>>>(X, Y, dr, w, partial);

  const double inv_total = 1.0 / ((double)maps * (double)OUT_W * (double)OUT_W);
  ssim_finalize_kernel<<<1, 256, 0, stream>>>(partial, nblocks, inv_total, out);
}